// Model_39676907884064
// MI455X (gfx1250) — compile-verified
//
#include <hip/hip_runtime.h>
#include <hip/hip_bf16.h>
#include <math.h>

typedef float v2f __attribute__((ext_vector_type(2)));
typedef float v4f __attribute__((ext_vector_type(4)));
typedef float v8f __attribute__((ext_vector_type(8)));

#define H 1024
#define S 48
#define NHD 32           // NH == DH == 32
#define ROW_REPS 1536    // S * NH = 48*32 copies of the 1024-float tile per output row
#define ROW_ELEMS 1572864  // ROW_REPS * 1024

// ---------------------------------------------------------------------------
// Kernel 1: Y[48,1024] = X[48,1024] @ W^T[1024,1024] + b   via V_WMMA_F32_16X16X4_F32
// One wave (32 lanes) computes one 16x16 tile of Y, accumulating K=1024 in steps of 4.
// A (16x4 f32) layout: lanes 0-15 -> {A[l,k0],A[l,k0+1]}, lanes 16-31 -> {A[l,k0+2],A[l,k0+3]}
// B (4x16 f32) layout: B[k,n] = W[n0+n, k]; lane n<16 -> {B[0,n],B[1,n]}, lane n+16 -> {B[2,n],B[3,n]}
// C/D (16x16 f32): acc[i] = C[i + 8*(lane>=16), lane&15]
// ---------------------------------------------------------------------------
__global__ __launch_bounds__(32) void proj_wmma(const float* __restrict__ X,
                                                const float* __restrict__ W,
                                                const float* __restrict__ bias,
                                                float* __restrict__ Y) {
  const int lane = threadIdx.x;
  const int m0 = blockIdx.x * 16;   // 0,16,32   (M = 48)
  const int n0 = blockIdx.y * 16;   // 0..1008   (N = 1024)
  const int half = lane >> 4;       // 0 or 1
  const int l = lane & 15;

  const float* arow = X + (size_t)(m0 + l) * H + half * 2;  // A[l, k0 + 2*half .. +1]
  const float* brow = W + (size_t)(n0 + l) * H + half * 2;  // B[k0+2*half.., n=l]

  v8f acc = {};
#pragma unroll 4
  for (int k0 = 0; k0 < H; k0 += 4) {
    v2f a = *(const v2f*)(arow + k0);
    v2f b = *(const v2f*)(brow + k0);
    acc = __builtin_amdgcn_wmma_f32_16x16x4_f32(false, a, false, b,
                                                (short)0, acc, false, false);
  }

  const float bn = bias[n0 + l];
#pragma unroll
  for (int i = 0; i < 8; ++i) {
    int m = i + 8 * half;
    Y[(size_t)(m0 + m) * H + (n0 + l)] = acc[i] + bn;
  }
}

// ---------------------------------------------------------------------------
// Kernel 2: per-s attention core. One wave per s; lane b owns head-row b.
//   qh[n,d] = qp[s,n*32+d] / 32      (inv_sqrt_h = 1/sqrt(1024))
//   _sq[b]  = sum_d qh[b,d]^2
//   _sk[c,b] = (1/32) * sum_n kh[n,c]*qh[n,b]
//   P[b,c]  = softmax_c( |_sk[c,b] - _sq[b]| * 0.2 )
//   O[b,d]  = sum_c P[b,c] * vh[c,d]
// ---------------------------------------------------------------------------
__global__ __launch_bounds__(32) void attn_core(const float* __restrict__ qp,
                                                const float* __restrict__ kp,
                                                const float* __restrict__ vp,
                                                float* __restrict__ O) {
  __shared__ float qs[H];
  __shared__ float ks[H];
  __shared__ float vs[H];
  const int s = blockIdx.x;
  const int b = threadIdx.x;        // 0..31
  const float inv = 1.0f / 32.0f;   // 1/sqrt(H)

  // cooperative, coalesced stage of this s's projected rows
  for (int t = 0; t < NHD; ++t) {
    int idx = t * 32 + b;
    qs[idx] = qp[(size_t)s * H + idx] * inv;   // pre-scaled qh
    ks[idx] = kp[(size_t)s * H + idx];
    vs[idx] = vp[(size_t)s * H + idx];
  }
  __syncthreads();

  // _sq[b]
  float sq = 0.f;
#pragma unroll
  for (int d = 0; d < NHD; ++d) { float x = qs[b * NHD + d]; sq = fmaf(x, x, sq); }

  // _sk[:, b]
  float skc[NHD];
  for (int c = 0; c < NHD; ++c) {
    float acc = 0.f;
#pragma unroll
    for (int n = 0; n < NHD; ++n) acc = fmaf(ks[n * NHD + c], qs[n * NHD + b], acc);
    skc[c] = acc * inv;
  }

  // softmax over c of |skc[c] - sq| * 0.2
  float t[NHD];
  float mx = -1e30f;
#pragma unroll
  for (int c = 0; c < NHD; ++c) { t[c] = fabsf(skc[c] - sq) * 0.2f; mx = fmaxf(mx, t[c]); }
  float sum = 0.f;
#pragma unroll
  for (int c = 0; c < NHD; ++c) { t[c] = __expf(t[c] - mx); sum += t[c]; }
  const float rs = 1.0f / sum;

  // O[b, :] = (P[b,:] @ vh) * rs
  for (int d = 0; d < NHD; ++d) {
    float acc = 0.f;
#pragma unroll
    for (int c = 0; c < NHD; ++c) acc = fmaf(t[c], vs[c * NHD + d], acc);
    O[(size_t)s * H + b * NHD + d] = acc * rs;
  }
}

// ---------------------------------------------------------------------------
// Kernel 3: broadcast-store. Output row j = 1536 contiguous copies of O[j] (1024 f32).
// 302 MB total -> bandwidth bound; use coalesced b128 non-temporal stores
// (output exceeds the 192 MB L2, so NT streaming is the right policy).
// grid = (96, 48), block = 256: each block writes 16 copies; each thread owns
// one v4f of the tile and stores it 16 times.
// ---------------------------------------------------------------------------
__global__ __launch_bounds__(256) void bcast_store(const float* __restrict__ O,
                                                   float* __restrict__ out) {
  const int j   = blockIdx.y;        // 0..47
  const int blk = blockIdx.x;        // 0..95
  const int t   = threadIdx.x;       // 0..255

  const v4f v = *(const v4f*)(O + (size_t)j * H + t * 4);
  float* base = out + (size_t)j * ROW_ELEMS + (size_t)blk * 16 * H + t * 4;
#pragma unroll
  for (int r = 0; r < 16; ++r) {
    __builtin_nontemporal_store(v, (v4f*)(base + (size_t)r * H));
  }
}

// ---------------------------------------------------------------------------
extern "C" void kernel_launch(void* const* d_in, const int* in_sizes, int n_in,
                              void* d_out, int out_size, void* d_ws, size_t ws_size,
                              hipStream_t stream) {
  const float* q  = (const float*)d_in[0];
  const float* k  = (const float*)d_in[1];
  const float* v  = (const float*)d_in[2];
  const float* Wq = (const float*)d_in[3];
  const float* bq = (const float*)d_in[4];
  const float* Wk = (const float*)d_in[5];
  const float* bk = (const float*)d_in[6];
  const float* Wv = (const float*)d_in[7];
  const float* bv = (const float*)d_in[8];
  float* out = (float*)d_out;

  float* ws = (float*)d_ws;
  float* qp = ws;                 // [48,1024]
  float* kp = ws + S * H;         // [48,1024]
  float* vp = ws + 2 * S * H;     // [48,1024]
  float* O  = ws + 3 * S * H;     // [48,1024]

  dim3 pg(3, H / 16);  // 3 M-tiles x 64 N-tiles
  proj_wmma<<<pg, 32, 0, stream>>>(q, Wq, bq, qp);
  proj_wmma<<<pg, 32, 0, stream>>>(k, Wk, bk, kp);
  proj_wmma<<<pg, 32, 0, stream>>>(v, Wv, bv, vp);

  attn_core<<<S, 32, 0, stream>>>(qp, kp, vp, O);

  bcast_store<<<dim3(ROW_REPS / 16, S), 256, 0, stream>>>(O, out);
}